// AKFVB_61246233641007
// MI455X (gfx1250) — compile-verified
//
#include <hip/hip_runtime.h>
#include <math.h>

// Kalman filter on MI455X (gfx1250): FP32 WMMA (v_wmma_f32_16x16x4_f32) for the
// 64x64 covariance recursion, Gauss-Jordan inverse in LDS, one block per batch elem.

typedef float v2f __attribute__((ext_vector_type(2)));
typedef float v8f __attribute__((ext_vector_type(8)));

#define LSEQ  256
#define BATCH 512
#define UDIM  32
#define NDIM  64
#define NT    256   // 8 wave32s per block

__device__ __forceinline__ float softplus_f(float x) {
  return (x > 20.f) ? x : log1pf(expf(x));
}

// ---------------------------------------------------------------------------
// Pre-kernel: build A^T, H^T, Qdiag=softplus(logsigma)^2, Rdiag=softplus(logdelta)^2
// into workspace so main-kernel WMMA right-operand loads are row-major/coalesced.
// ws layout (floats): [0,4096) At | [4096,8192) Ht | [8192,8256) qd | [8256,8320) rd
// ---------------------------------------------------------------------------
__global__ void kf_precompute(const float* __restrict__ A,
                              const float* __restrict__ H,
                              const float* __restrict__ logdelta,
                              const float* __restrict__ logsigma,
                              float* __restrict__ ws) {
  int tid = blockIdx.x * blockDim.x + threadIdx.x;
  if (tid < NDIM * NDIM) {
    int i = tid >> 6, j = tid & 63;
    ws[j * NDIM + i]               = A[tid];  // At
    ws[NDIM * NDIM + j * NDIM + i] = H[tid];  // Ht
  }
  if (tid < NDIM) {
    float sq = softplus_f(logsigma[tid]);
    float sd = softplus_f(logdelta[tid]);
    ws[2 * NDIM * NDIM + tid]        = sq * sq;  // Q diag (process noise)
    ws[2 * NDIM * NDIM + NDIM + tid] = sd * sd;  // R diag (obs noise)
  }
}

// ---------------------------------------------------------------------------
// 64x64x64 FP32 matmul: dst = L * R, via V_WMMA_F32_16X16X4_F32.
// 8 waves; wave w owns output tiles (tm=w>>2, tn=w&3) and (tm+2, tn) -> shares
// the B fragment. Compute fully into registers, barrier, then write: dst may
// alias L or R. All pointers generic (LDS or global); inlining recovers ds/global.
//
// f32 A-frag (16x4, ISA 7.12.2): lanes 0-15 hold K={k0,k0+1}, lanes 16-31 K={k0+2,k0+3}.
// f32 B-frag (4x16): VGPR pair analogous (rows split across lane halves).
// f32 C/D  (16x16): VGPR r = rows r / r+8 across lane halves.
// ---------------------------------------------------------------------------
__device__ __forceinline__ void mm64(float* dst, const float* L, const float* R, int tid) {
  const int wave = tid >> 5;
  const int lane = tid & 31;
  const int half = lane >> 4;       // 0: lanes 0-15, 1: lanes 16-31
  const int l15  = lane & 15;
  const int tn   = wave & 3;
  const int tm0  = wave >> 2;       // 0 or 1
  const int tm1  = tm0 + 2;         // 2 or 3

  v8f acc0 = {};
  v8f acc1 = {};

  __syncthreads();                  // sources ready

  const int bcol  = tn * 16 + l15;
  const int arow0 = tm0 * 16 + l15;
  const int arow1 = tm1 * 16 + l15;

#pragma unroll
  for (int k0 = 0; k0 < NDIM; k0 += 4) {
    const int kk = k0 + half * 2;
    v2f a0, a1, bb;
    a0.x = L[arow0 * NDIM + kk];
    a0.y = L[arow0 * NDIM + kk + 1];
    a1.x = L[arow1 * NDIM + kk];
    a1.y = L[arow1 * NDIM + kk + 1];
    bb.x = R[kk * NDIM + bcol];
    bb.y = R[(kk + 1) * NDIM + bcol];
    acc0 = __builtin_amdgcn_wmma_f32_16x16x4_f32(false, a0, false, bb, (short)0, acc0,
                                                 false, false);
    acc1 = __builtin_amdgcn_wmma_f32_16x16x4_f32(false, a1, false, bb, (short)0, acc1,
                                                 false, false);
  }

  __syncthreads();                  // all reads done -> safe to overwrite aliased dst

#pragma unroll
  for (int r = 0; r < 8; ++r) {
    dst[(tm0 * 16 + half * 8 + r) * NDIM + tn * 16 + l15] = acc0[r];
    dst[(tm1 * 16 + half * 8 + r) * NDIM + tn * 16 + l15] = acc1[r];
  }

  __syncthreads();                  // dst visible
}

// ---------------------------------------------------------------------------
// Main kernel: one block per batch element, sequential over L timesteps.
// ---------------------------------------------------------------------------
__global__ __launch_bounds__(NT)
void kf_kernel(const float* __restrict__ ext_u,  // [L,B,U]
               const float* __restrict__ obs_y,  // [L,B,N]
               const float* __restrict__ mu0,    // [B,N]
               const float* __restrict__ sig0,   // [B,N]
               const float* __restrict__ A,      // [N,N]
               const float* __restrict__ Bm,     // [N,U]
               const float* __restrict__ H,      // [N,N]
               const float* __restrict__ At,     // [N,N] (ws)
               const float* __restrict__ Ht,     // [N,N] (ws)
               const float* __restrict__ qd,     // [N]   (ws)
               const float* __restrict__ rd,     // [N]   (ws)
               float* __restrict__ out_mu,       // [L,B,N]
               float* __restrict__ out_ls) {     // [L,B,N]
  __shared__ float covS[NDIM * NDIM];   // persistent state / S / Sinv / G / cov_post
  __shared__ float Xs[NDIM * NDIM];     // scratch
  __shared__ float Ys[NDIM * NDIM];     // scratch (holds P through the step)
  __shared__ float fv[NDIM];
  __shared__ float muS[NDIM], mupredS[NDIM], innovS[NDIM], mupostS[NDIM];
  __shared__ float uS[UDIM], yS[NDIM];

  const int b   = blockIdx.x;
  const int tid = threadIdx.x;

  // init: cov = diag(sigma0^2), mu = mu0
  const float* mu0b = mu0 + (size_t)b * NDIM;
  const float* s0b  = sig0 + (size_t)b * NDIM;
  for (int idx = tid; idx < NDIM * NDIM; idx += NT) {
    int i = idx >> 6, j = idx & 63;
    float v = 0.f;
    if (i == j) { float s = s0b[i]; v = s * s; }
    covS[idx] = v;
  }
  if (tid < NDIM) muS[tid] = mu0b[tid];
  __syncthreads();

  for (int t = 0; t < LSEQ; ++t) {
    const float* up = ext_u + ((size_t)t * BATCH + b) * UDIM;
    const float* yp = obs_y + ((size_t)t * BATCH + b) * NDIM;
    if (tid < UDIM)                     uS[tid] = up[tid];
    else if (tid < UDIM + NDIM)         yS[tid - UDIM] = yp[tid - UDIM];
    __syncthreads();

    // mu_pred = A @ mu + Bm @ u
    if (tid < NDIM) {
      float acc = 0.f;
      for (int j = 0; j < NDIM; ++j) acc += A[tid * NDIM + j] * muS[j];
      for (int j = 0; j < UDIM; ++j) acc += Bm[tid * UDIM + j] * uS[j];
      mupredS[tid] = acc;
    }
    __syncthreads();

    // innov = y - H @ mu_pred
    if (tid < NDIM) {
      float acc = 0.f;
      for (int j = 0; j < NDIM; ++j) acc += H[tid * NDIM + j] * mupredS[j];
      innovS[tid] = yS[tid] - acc;
    }
    // (mm64's entry barrier orders this write before any later read)

    // ---- covariance chain (all 64^3 WMMA matmuls) ----
    mm64(Xs, A, covS, tid);                       // X = A @ cov        (cov dead)
    mm64(Ys, Xs, At, tid);                        // Y = X @ A^T  -> P
    if (tid < NDIM) Ys[tid * NDIM + tid] += qd[tid];   // P += Q
    mm64(Xs, H, Ys, tid);                         // X = H @ P    -> M
    mm64(covS, Xs, Ht, tid);                      // cov = M @ H^T -> S
    if (tid < NDIM) covS[tid * NDIM + tid] += rd[tid]; // S += R
    __syncthreads();

    // ---- in-place Gauss-Jordan inverse of S (SPD, no pivoting) ----
    for (int k = 0; k < NDIM; ++k) {
      float piv = covS[k * NDIM + k];
      float p   = 1.0f / piv;
      __syncthreads();
      if (tid < NDIM) {                           // scale pivot row
        if (tid != k) covS[k * NDIM + tid] *= p;
        else          covS[k * NDIM + k]   = p;
      } else if (tid < 2 * NDIM) {                // snapshot pivot column
        int i = tid - NDIM;
        fv[i] = (i == k) ? 0.f : covS[i * NDIM + k];
      }
      __syncthreads();
      for (int idx = tid; idx < NDIM * NDIM; idx += NT) {
        int i = idx >> 6, j = idx & 63;
        if (i == k) continue;
        float f = fv[i];
        if (j == k) covS[idx]  = -f * p;
        else        covS[idx] -= f * covS[k * NDIM + j];
      }
      __syncthreads();
    }
    // covS now holds Sinv

    mm64(Xs, Ys, Ht, tid);                        // X = P @ H^T       (M dead)
    mm64(Xs, Xs, covS, tid);                      // X = PHt @ Sinv -> K (aliased, safe)

    // mu_post = mu_pred + K @ innov
    if (tid < NDIM) {
      float acc = 0.f;
      for (int j = 0; j < NDIM; ++j) acc += Xs[tid * NDIM + j] * innovS[j];
      mupostS[tid] = mupredS[tid] + acc;
    }

    mm64(covS, Xs, H, tid);                       // cov = K @ H  -> G (Sinv dead)
    mm64(Xs, covS, Ys, tid);                      // X = G @ P         (K dead)

    // cov_post = P - G@P ; emit outputs
    for (int idx = tid; idx < NDIM * NDIM; idx += NT)
      covS[idx] = Ys[idx] - Xs[idx];
    __syncthreads();

    const size_t obase = ((size_t)t * BATCH + b) * NDIM;
    if (tid < NDIM) {
      float c = covS[tid * NDIM + tid];
      float s = sqrtf(c);
      float ls = s + logf(-expm1f(-s));           // inverse_softplus(sqrt(diag))
      out_mu[obase + tid] = mupostS[tid];
      out_ls[obase + tid] = ls;
      muS[tid] = mupostS[tid];
    }
    __syncthreads();
  }
}

// ---------------------------------------------------------------------------
extern "C" void kernel_launch(void* const* d_in, const int* in_sizes, int n_in,
                              void* d_out, int out_size, void* d_ws, size_t ws_size,
                              hipStream_t stream) {
  (void)in_sizes; (void)n_in; (void)out_size; (void)ws_size;
  const float* ext_u    = (const float*)d_in[0];
  const float* obs_y    = (const float*)d_in[1];
  const float* mu0      = (const float*)d_in[2];
  const float* sig0     = (const float*)d_in[3];
  const float* A        = (const float*)d_in[4];
  const float* Bm       = (const float*)d_in[5];
  const float* H        = (const float*)d_in[6];
  const float* logdelta = (const float*)d_in[7];
  const float* logsigma = (const float*)d_in[8];

  float* ws = (float*)d_ws;
  float* At = ws;
  float* Ht = ws + NDIM * NDIM;
  float* qd = ws + 2 * NDIM * NDIM;
  float* rd = qd + NDIM;

  kf_precompute<<<(NDIM * NDIM + 255) / 256, 256, 0, stream>>>(A, H, logdelta, logsigma, ws);

  float* out_mu = (float*)d_out;
  float* out_ls = out_mu + (size_t)LSEQ * BATCH * NDIM;
  kf_kernel<<<BATCH, NT, 0, stream>>>(ext_u, obs_y, mu0, sig0, A, Bm, H,
                                      At, Ht, qd, rd, out_mu, out_ls);
}